// BayesianAtlas_75840532513311
// MI455X (gfx1250) — compile-verified
//
#include <hip/hip_runtime.h>

typedef __attribute__((ext_vector_type(2))) float v2f;
typedef __attribute__((ext_vector_type(8))) float v8f;

#define NB    256      // batch
#define DG    16
#define LAT   10
#define NPTS  20000
#define TSTEP 10       // T-1
#define DT    0.1f

// ---------------------------------------------------------------------------
// Kernel 1: encoder conv stack.  One block per image; all intermediates in LDS.
// obs [B,2,64,64] -> feat [B,256]  (flatten of [16,4,4], channel-major)
// ---------------------------------------------------------------------------
__global__ void enc_conv_kernel(const float* __restrict__ obs,
                                const float* __restrict__ w1, const float* __restrict__ b1,
                                const float* __restrict__ w2, const float* __restrict__ b2,
                                const float* __restrict__ w3, const float* __restrict__ b3,
                                const float* __restrict__ w4, const float* __restrict__ b4,
                                float* __restrict__ feat) {
  __shared__ float l1[4 * 32 * 32];
  __shared__ float l2[8 * 16 * 16];
  __shared__ float l3[16 * 8 * 8];
  const int b = blockIdx.x, tid = threadIdx.x;
  const float* ob = obs + (size_t)b * 2 * 64 * 64;

  // conv1: [2,64,64] -> [4,32,32], tanh
  for (int i = tid; i < 4096; i += 256) {
    int co = i >> 10, y = (i >> 5) & 31, x = i & 31;
    float s = b1[co];
    for (int ci = 0; ci < 2; ++ci)
      for (int ky = 0; ky < 2; ++ky)
        for (int kx = 0; kx < 2; ++kx)
          s += ob[(ci * 64 + 2 * y + ky) * 64 + 2 * x + kx] *
               w1[((co * 2 + ci) * 2 + ky) * 2 + kx];
    l1[i] = tanhf(s);
  }
  __syncthreads();
  // conv2: [4,32,32] -> [8,16,16], tanh
  for (int i = tid; i < 2048; i += 256) {
    int co = i >> 8, y = (i >> 4) & 15, x = i & 15;
    float s = b2[co];
    for (int ci = 0; ci < 4; ++ci)
      for (int ky = 0; ky < 2; ++ky)
        for (int kx = 0; kx < 2; ++kx)
          s += l1[ci * 1024 + (2 * y + ky) * 32 + 2 * x + kx] *
               w2[((co * 4 + ci) * 2 + ky) * 2 + kx];
    l2[i] = tanhf(s);
  }
  __syncthreads();
  // conv3: [8,16,16] -> [16,8,8], tanh
  for (int i = tid; i < 1024; i += 256) {
    int co = i >> 6, y = (i >> 3) & 7, x = i & 7;
    float s = b3[co];
    for (int ci = 0; ci < 8; ++ci)
      for (int ky = 0; ky < 2; ++ky)
        for (int kx = 0; kx < 2; ++kx)
          s += l2[ci * 256 + (2 * y + ky) * 16 + 2 * x + kx] *
               w3[((co * 8 + ci) * 2 + ky) * 2 + kx];
    l3[i] = tanhf(s);
  }
  __syncthreads();
  // conv4: [16,8,8] -> [16,4,4], tanh; write flattened feature vector
  {
    int i = tid;  // exactly 256 outputs
    int co = i >> 4, y = (i >> 2) & 3, x = i & 3;
    float s = b4[co];
    for (int ci = 0; ci < 16; ++ci)
      for (int ky = 0; ky < 2; ++ky)
        for (int kx = 0; kx < 2; ++kx)
          s += l3[ci * 64 + (2 * y + ky) * 8 + 2 * x + kx] *
               w4[((co * 16 + ci) * 2 + ky) * 2 + kx];
    feat[b * 256 + i] = tanhf(s);
  }
}

// ---------------------------------------------------------------------------
// Kernel 2: encoder linear via exact-fp32 WMMA.  z = feat @ W^T + bias
// feat [256,256], W [10,256] -> z [256,10].  M=256 (16 tiles), N=16 (10 used),
// K=256 (64 steps of V_WMMA_F32_16X16X4_F32).  1 wave per M-tile.
// ---------------------------------------------------------------------------
__global__ void enc_linear_wmma(const float* __restrict__ feat,
                                const float* __restrict__ W,
                                const float* __restrict__ bias,
                                float* __restrict__ z) {
  const int lane = threadIdx.x & 31;
  const int wv   = threadIdx.x >> 5;
  const int mt   = blockIdx.x * 8 + wv;       // 0..15
  const int m0   = mt * 16;
  const int half = lane >> 4;                 // selects K pair within step
  const int mrow = lane & 15;
  const int n    = mrow;                      // B-fragment column index
  const float nv = (n < LAT) ? 1.0f : 0.0f;   // zero-pad N beyond 10
  const float* Wn = W + (n < LAT ? n : 0) * 256;
  const float* Am = feat + (m0 + mrow) * 256;

  v8f acc = {};
  for (int kk = 0; kk < 256; kk += 4) {
    int k = kk + half * 2;
    v2f a  = *(const v2f*)(Am + k);
    v2f bf = *(const v2f*)(Wn + k);
    bf.x *= nv;  bf.y *= nv;
    acc = __builtin_amdgcn_wmma_f32_16x16x4_f32(false, a, false, bf,
                                                (short)0, acc, false, false);
  }
  if (n < LAT) {
    float bb = bias[n];
    for (int j = 0; j < 8; ++j) {
      int m = m0 + half * 8 + j;              // D layout: M = j (+8 for hi lanes)
      z[m * LAT + n] = acc[j] + bb;
    }
  }
}

// ---------------------------------------------------------------------------
// Kernel 3: decoder linear via fp32 WMMA with the time-scale fused into A.
// row r = t*256 + b, A[r,k] = (t+1)*DT * z[b,k];  W [64,10] (bias-free)
// h0[r, 0..63] = tanh(A @ W^T).  M=2560 (160 tiles), N=64 (4 tiles), K=10->12.
// ---------------------------------------------------------------------------
__global__ void dec_linear_wmma(const float* __restrict__ z,
                                const float* __restrict__ W,
                                float* __restrict__ h0) {
  const int lane = threadIdx.x & 31;
  const int wv   = threadIdx.x >> 5;
  const int mt   = blockIdx.x * 8 + wv;       // 0..159
  const int m0   = mt * 16;
  const int half = lane >> 4;
  const int mrow = lane & 15;
  const int m    = m0 + mrow;
  const int tt   = m >> 8;                    // time index 0..9
  const int bb   = m & 255;                   // batch index
  const float scale = (float)(tt + 1) * DT;

  v2f a[3], bf[3][4];
  for (int ks = 0; ks < 3; ++ks) {
    int k = ks * 4 + half * 2;
    int k0 = (k     < LAT) ? k     : 0;  float m0f = (k     < LAT) ? scale : 0.0f;
    int k1 = (k + 1 < LAT) ? k + 1 : 0;  float m1f = (k + 1 < LAT) ? scale : 0.0f;
    a[ks].x = z[bb * LAT + k0] * m0f;
    a[ks].y = z[bb * LAT + k1] * m1f;
    for (int nt = 0; nt < 4; ++nt) {
      int n = nt * 16 + mrow;               // 0..63, always valid
      float v0 = (k     < LAT) ? 1.0f : 0.0f;
      float v1 = (k + 1 < LAT) ? 1.0f : 0.0f;
      bf[ks][nt].x = W[n * LAT + k0] * v0;
      bf[ks][nt].y = W[n * LAT + k1] * v1;
    }
  }
  v8f acc[4] = {};
  for (int ks = 0; ks < 3; ++ks)
    for (int nt = 0; nt < 4; ++nt)
      acc[nt] = __builtin_amdgcn_wmma_f32_16x16x4_f32(false, a[ks], false, bf[ks][nt],
                                                      (short)0, acc[nt], false, false);
  for (int nt = 0; nt < 4; ++nt)
    for (int j = 0; j < 8; ++j) {
      int mm = m0 + half * 8 + j;
      int n  = nt * 16 + mrow;
      h0[mm * 64 + n] = tanhf(acc[nt][j]);
    }
}

// ---------------------------------------------------------------------------
// Kernel 4: decoder convT chain.  8 rows per block through LDS.
// h0 [r,16,2,2] -> convT(16->8)+tanh -> convT(8->4)+tanh -> convT(4->2)
// Output written pre-transposed as vel[r][H][W][c]  (r = t*256 + b).
// convT(k=2,s=2): out[o,y,x] = sum_i in[i,y/2,x/2] * w[i,o,y&1,x&1]
// ---------------------------------------------------------------------------
__global__ void dec_convt(const float* __restrict__ h0,
                          const float* __restrict__ w1,   // [16,8,2,2] IOHW
                          const float* __restrict__ w2,   // [8,4,2,2]
                          const float* __restrict__ w3,   // [4,2,2,2]
                          float* __restrict__ vel) {      // [2560,16,16,2]
  __shared__ float h0s[8 * 64];
  __shared__ float h1[8 * 128];
  __shared__ float h2[8 * 256];
  const int tid = threadIdx.x;
  const int r0  = blockIdx.x * 8;

  for (int i = tid; i < 512; i += 256) h0s[i] = h0[r0 * 64 + i];
  __syncthreads();
  // convT1: [16,2,2] -> [8,4,4], tanh
  for (int i = tid; i < 1024; i += 256) {
    int row = i >> 7, rem = i & 127;
    int co = rem >> 4, y = (rem >> 2) & 3, x = rem & 3;
    float s = 0.0f;
    for (int ci = 0; ci < 16; ++ci)
      s += h0s[row * 64 + ci * 4 + (y >> 1) * 2 + (x >> 1)] *
           w1[((ci * 8 + co) * 2 + (y & 1)) * 2 + (x & 1)];
    h1[row * 128 + co * 16 + y * 4 + x] = tanhf(s);
  }
  __syncthreads();
  // convT2: [8,4,4] -> [4,8,8], tanh
  for (int i = tid; i < 2048; i += 256) {
    int row = i >> 8, rem = i & 255;
    int co = rem >> 6, y = (rem >> 3) & 7, x = rem & 7;
    float s = 0.0f;
    for (int ci = 0; ci < 8; ++ci)
      s += h1[row * 128 + ci * 16 + (y >> 1) * 4 + (x >> 1)] *
           w2[((ci * 4 + co) * 2 + (y & 1)) * 2 + (x & 1)];
    h2[row * 256 + co * 64 + y * 8 + x] = tanhf(s);
  }
  __syncthreads();
  // convT3: [4,8,8] -> [2,16,16]; store channel-interleaved [H][W][c]
  for (int i = tid; i < 4096; i += 256) {
    int row = i >> 9, rem = i & 511;
    int co = rem >> 8, y = (rem >> 4) & 15, x = rem & 15;
    float s = 0.0f;
    for (int ci = 0; ci < 4; ++ci)
      s += h2[row * 256 + ci * 64 + (y >> 1) * 8 + (x >> 1)] *
           w3[((ci * 2 + co) * 2 + (y & 1)) * 2 + (x & 1)];
    vel[(size_t)(r0 + row) * 512 + (y * 16 + x) * 2 + co] = s;
  }
}

// ---------------------------------------------------------------------------
// Kernel 5: fused 10-step Euler integration.  One block per (point-chunk,
// batch).  All 10 velocity grids for the batch live in 20KB of LDS; each
// point stays in registers across the whole time loop.
// ---------------------------------------------------------------------------
__global__ void integrate_kernel(const float* __restrict__ tpl,   // [20000,2]
                                 const float* __restrict__ vel,   // [t*256+b][512]
                                 float* __restrict__ out) {       // [256,20000,2]
  __shared__ float lv[TSTEP * 512];
  const int tid = threadIdx.x;
  const int b   = blockIdx.y;
  const int pc  = blockIdx.x;

  for (int i = tid; i < TSTEP * 512; i += 256) {
    int t = i >> 9, s = i & 511;
    lv[i] = vel[(size_t)(t * NB + b) * 512 + s];
  }
  __syncthreads();

  const float2* tp = (const float2*)tpl;
  float2* op = (float2*)(out + (size_t)b * NPTS * 2);
  const int pend = (pc + 1) * 2500;   // 8 chunks * 2500 = NPTS exactly

  for (int p = pc * 2500 + tid; p < pend; p += 256) {
    __builtin_prefetch(&tp[p + 256], 0, 1);   // speculative -> global_prefetch_b8
    float2 P = tp[p];
    float px = P.x, py = P.y;
#pragma unroll
    for (int t = 0; t < TSTEP; ++t) {
      float u = (px + 2.5f) * 3.0f;           // (x - lo)/(hi-lo) * (DG-1)
      float v = (py + 2.5f) * 3.0f;
      float uf = fminf(fmaxf(floorf(u), 0.0f), 15.0f);
      float vf = fminf(fmaxf(floorf(v), 0.0f), 15.0f);
      float u2 = fminf(uf + 1.0f, 15.0f);
      float v2 = fminf(vf + 1.0f, 15.0f);
      float fu = u - uf, fv = v - vf;
      float gu = uf + 1.0f - u, gv = vf + 1.0f - v;   // matches reference exactly
      int u1i = (int)uf, v1i = (int)vf, u2i = (int)u2, v2i = (int)v2;
      const float* g = lv + t * 512;
      float2 q11 = *(const float2*)(g + (u1i * 16 + v1i) * 2);
      float2 q12 = *(const float2*)(g + (u1i * 16 + v2i) * 2);
      float2 q21 = *(const float2*)(g + (u2i * 16 + v1i) * 2);
      float2 q22 = *(const float2*)(g + (u2i * 16 + v2i) * 2);
      float w11 = gu * gv, w12 = gu * fv, w21 = fu * gv, w22 = fu * fv;
      px += DT * (q11.x * w11 + q12.x * w12 + q21.x * w21 + q22.x * w22);
      py += DT * (q11.y * w11 + q12.y * w12 + q21.y * w21 + q22.y * w22);
    }
    float2 R; R.x = px; R.y = py;
    op[p] = R;
  }
}

// ---------------------------------------------------------------------------
extern "C" void kernel_launch(void* const* d_in, const int* in_sizes, int n_in,
                              void* d_out, int out_size, void* d_ws, size_t ws_size,
                              hipStream_t stream) {
  const float* obs = (const float*)d_in[0];
  const float* tpl = (const float*)d_in[1];
  const float* ew1 = (const float*)d_in[2];
  const float* eb1 = (const float*)d_in[3];
  const float* ew2 = (const float*)d_in[4];
  const float* eb2 = (const float*)d_in[5];
  const float* ew3 = (const float*)d_in[6];
  const float* eb3 = (const float*)d_in[7];
  const float* ew4 = (const float*)d_in[8];
  const float* eb4 = (const float*)d_in[9];
  const float* elw = (const float*)d_in[10];   // [10,256]
  const float* elb = (const float*)d_in[11];   // [10]
  const float* dlw = (const float*)d_in[12];   // [64,10]
  const float* dw1 = (const float*)d_in[13];   // [16,8,2,2]
  const float* dw2 = (const float*)d_in[14];   // [8,4,2,2]
  const float* dw3 = (const float*)d_in[15];   // [4,2,2,2]

  // workspace layout (floats):  feat 65536 | z 2560 | h0 163840 | vel 1310720
  float* feat = (float*)d_ws;
  float* z    = feat + 65536;
  float* h0   = z + 2560;
  float* vel  = h0 + 2560 * 64;

  enc_conv_kernel<<<NB, 256, 0, stream>>>(obs, ew1, eb1, ew2, eb2, ew3, eb3,
                                          ew4, eb4, feat);
  enc_linear_wmma<<<2, 256, 0, stream>>>(feat, elw, elb, z);
  dec_linear_wmma<<<20, 256, 0, stream>>>(z, dlw, h0);
  dec_convt<<<320, 256, 0, stream>>>(h0, dw1, dw2, dw3, vel);
  integrate_kernel<<<dim3(8, NB), 256, 0, stream>>>(tpl, vel, (float*)d_out);
}